// GNNModel_11287174054179
// MI455X (gfx1250) — compile-verified
//
#include <hip/hip_runtime.h>
#include <hip/hip_bf16.h>
#include <cstdint>

typedef float v2f __attribute__((ext_vector_type(2)));
typedef float v8f __attribute__((ext_vector_type(8)));

#define HDIM   128
#define GRAPHS 64
#define BN_EPS 1e-5f

static __device__ __forceinline__ void atomAddF(float* p, float v) {
    unsafeAtomicAdd(p, v);   // hardware global_atomic_add_f32 / ds_add_f32
}

// ---------------------------------------------------------------- utilities
__global__ void k_fill(float* __restrict__ p, float v, long n) {
    long i = (long)blockIdx.x * blockDim.x + threadIdx.x;
    if (i < n) p[i] = v;
}

__global__ void k_deg_accum(const int* __restrict__ col, float* __restrict__ deg, int E) {
    int i = blockIdx.x * blockDim.x + threadIdx.x;
    if (i < E) atomAddF(&deg[col[i]], 1.0f);
}

__global__ void k_rsqrt_inplace(float* __restrict__ p, int n) {
    int i = blockIdx.x * blockDim.x + threadIdx.x;
    if (i < n) p[i] = rsqrtf(p[i]);   // deg >= 1 always (self loop)
}

// ---------------------------------------------------------------- fp32 WMMA GEMM: Out(N x128) = A(N x128) @ W(128x128)
// 4 waves/block, one 16-row tile per wave. W staged in LDS.
__global__ void __launch_bounds__(128) k_gemm128(const float* __restrict__ A,
                                                 const float* __restrict__ W,
                                                 float* __restrict__ Out, int nrows) {
    __shared__ float sW[HDIM * HDIM];           // 64 KB
    const int t = threadIdx.x;
    for (int i = t * 4; i < HDIM * HDIM; i += 128 * 4)
        *(float4*)&sW[i] = *(const float4*)&W[i];
    __syncthreads();

    const int wave  = t >> 5;
    const int lane  = t & 31;
    const int m     = lane & 15;
    const int khalf = (lane >> 4) << 1;         // lanes 0-15 -> K{0,1}; 16-31 -> K{2,3}
    const int rowTile = blockIdx.x * 4 + wave;
    int row = rowTile * 16 + m;
    if (row >= nrows) row = nrows - 1;          // clamp; OOB rows masked at store
    const float* arow = A + (size_t)row * HDIM;

    v2f a[32];
#pragma unroll
    for (int s = 0; s < 32; ++s) {
        int k  = (s << 2) + khalf;
        a[s].x = arow[k];
        a[s].y = arow[k + 1];
    }

    const int n0    = lane & 15;
    const int mbase = rowTile * 16 + ((lane >> 4) << 3);
    for (int nt = 0; nt < 8; ++nt) {
        v8f c = {0.f, 0.f, 0.f, 0.f, 0.f, 0.f, 0.f, 0.f};
#pragma unroll
        for (int s = 0; s < 32; ++s) {
            int k = (s << 2) + khalf;
            v2f b;
            b.x = sW[k * HDIM + nt * 16 + n0];
            b.y = sW[(k + 1) * HDIM + nt * 16 + n0];
            c = __builtin_amdgcn_wmma_f32_16x16x4_f32(false, a[s], false, b,
                                                      (short)0, c, false, false);
        }
        const int colo = nt * 16 + n0;
#pragma unroll
        for (int r = 0; r < 8; ++r) {
            int orow = mbase + r;
            if (orow < nrows) Out[(size_t)orow * HDIM + colo] = c[r];
        }
    }
}

// ---------------------------------------------------------------- self-loop init: out[i] = dinv[i]^2 * hw[i]
// Replaces zero-fill + self-loop atomics with coalesced useful stores.
__global__ void k_selfinit(const float* __restrict__ hw, const float* __restrict__ dinv,
                           float* __restrict__ out, long n4) {
    long i = (long)blockIdx.x * blockDim.x + threadIdx.x;
    if (i >= n4) return;
    long rowi = i >> 5;                          // 32 float4 per 128-wide row
    float d = dinv[rowi];
    d = d * d;                                   // dinv[i]*dinv[i] = 1/deg[i]
    float4 v = ((const float4*)hw)[i];
    v.x *= d; v.y *= d; v.z *= d; v.w *= d;
    ((float4*)out)[i] = v;
}

// ---------------------------------------------------------------- edge scatter: out[col] += dinv[row]*dinv[col]*hw[row]
// one wave per edge; each lane handles 4 features.
__global__ void k_scatter(const float* __restrict__ hw, const int* __restrict__ rowI,
                          const int* __restrict__ colI, const float* __restrict__ dinv,
                          float* __restrict__ out, int E) {
    long gid = (long)blockIdx.x * blockDim.x + threadIdx.x;
    long e   = gid >> 5;
    int lane = (int)(gid & 31);
    if (e >= E) return;
    int r = rowI[e];
    int c = colI[e];
    float nrm = dinv[r] * dinv[c];
    float4 v  = *(const float4*)&hw[(size_t)r * HDIM + lane * 4];
    float* dst = &out[(size_t)c * HDIM + lane * 4];
    atomAddF(dst + 0, nrm * v.x);
    atomAddF(dst + 1, nrm * v.y);
    atomAddF(dst + 2, nrm * v.z);
    atomAddF(dst + 3, nrm * v.w);
}

// ---------------------------------------------------------------- per-column sum / sumsq
__global__ void k_stats(const float* __restrict__ h, float* __restrict__ sums,
                        float* __restrict__ sumsq, int nrows) {
    const int f   = threadIdx.x & 127;
    const int sub = threadIdx.x >> 7;           // 2 rows in flight per block
    float s = 0.f, s2 = 0.f;
    for (int i = blockIdx.x * 2 + sub; i < nrows; i += gridDim.x * 2) {
        float v = h[(size_t)i * HDIM + f];
        s += v; s2 += v * v;
    }
    __shared__ float ls[256], ls2[256];
    ls[threadIdx.x] = s; ls2[threadIdx.x] = s2;
    __syncthreads();
    if (sub == 0) {
        s  += ls[threadIdx.x + 128];
        s2 += ls2[threadIdx.x + 128];
        atomAddF(&sums[f], s);
        atomAddF(&sumsq[f], s2);
    }
}

__global__ void k_bn_finalize(const float* __restrict__ sums, const float* __restrict__ sumsq,
                              const float* __restrict__ g, const float* __restrict__ beta,
                              float* __restrict__ scale, float* __restrict__ shift, float invN) {
    int f = threadIdx.x;
    float mu  = sums[f] * invN;
    float var = sumsq[f] * invN - mu * mu;
    float s   = g[f] * rsqrtf(var + BN_EPS);
    scale[f] = s;
    shift[f] = beta[f] - mu * s;
}

__global__ void k_bnrelu(float* __restrict__ h, const float* __restrict__ scale,
                         const float* __restrict__ shift, long n4) {
    long i = (long)blockIdx.x * blockDim.x + threadIdx.x;
    if (i >= n4) return;
    float4 v = ((float4*)h)[i];
    int f = (int)((i & 31) << 2);               // 32 float4 per 128-wide row
    v.x = fmaxf(fmaf(v.x, scale[f + 0], shift[f + 0]), 0.f);
    v.y = fmaxf(fmaf(v.y, scale[f + 1], shift[f + 1]), 0.f);
    v.z = fmaxf(fmaf(v.z, scale[f + 2], shift[f + 2]), 0.f);
    v.w = fmaxf(fmaf(v.w, scale[f + 3], shift[f + 3]), 0.f);
    ((float4*)h)[i] = v;
}

// ---------------------------------------------------------------- LDS-staged global mean pool (sums + counts)
__global__ void k_pool(const float* __restrict__ h, const int* __restrict__ batch,
                       float* __restrict__ pool, float* __restrict__ cnt, int nrows) {
    __shared__ float lp[GRAPHS * HDIM];         // 32 KB
    __shared__ float lc[GRAPHS];
    for (int i = threadIdx.x; i < GRAPHS * HDIM; i += blockDim.x) lp[i] = 0.f;
    if (threadIdx.x < GRAPHS) lc[threadIdx.x] = 0.f;
    __syncthreads();
    const int f   = threadIdx.x & 127;
    const int sub = threadIdx.x >> 7;
    for (int i = blockIdx.x * 2 + sub; i < nrows; i += gridDim.x * 2) {
        int b = batch[i];
        atomAddF(&lp[b * HDIM + f], h[(size_t)i * HDIM + f]);
        if (f == 0) atomAddF(&lc[b], 1.0f);
    }
    __syncthreads();
    for (int i = threadIdx.x; i < GRAPHS * HDIM; i += blockDim.x) {
        float v = lp[i];
        if (v != 0.f) atomAddF(&pool[i], v);
    }
    if (threadIdx.x < GRAPHS) {
        float v = lc[threadIdx.x];
        if (v != 0.f) atomAddF(&cnt[threadIdx.x], v);
    }
}

// ---------------------------------------------------------------- final FC: out(64x128) = (pool/cnt) @ fc_w + fc_b  (WMMA)
__global__ void __launch_bounds__(128) k_fc(const float* __restrict__ pool, const float* __restrict__ cnt,
                                            const float* __restrict__ Wf, const float* __restrict__ bf,
                                            float* __restrict__ out) {
    __shared__ float sW[HDIM * HDIM];
    const int t = threadIdx.x;
    for (int i = t * 4; i < HDIM * HDIM; i += 128 * 4)
        *(float4*)&sW[i] = *(const float4*)&Wf[i];
    __syncthreads();

    const int wave  = t >> 5, lane = t & 31;
    const int m     = lane & 15;
    const int khalf = (lane >> 4) << 1;
    const int row   = wave * 16 + m;            // 0..63
    const float inv = 1.0f / fmaxf(cnt[row], 1.0f);

    v2f a[32];
#pragma unroll
    for (int s = 0; s < 32; ++s) {
        int k  = (s << 2) + khalf;
        a[s].x = pool[row * HDIM + k] * inv;
        a[s].y = pool[row * HDIM + k + 1] * inv;
    }
    const int n0 = lane & 15;
    const int mb = wave * 16 + ((lane >> 4) << 3);
    for (int nt = 0; nt < 8; ++nt) {
        v8f c = {0.f, 0.f, 0.f, 0.f, 0.f, 0.f, 0.f, 0.f};
#pragma unroll
        for (int s = 0; s < 32; ++s) {
            int k = (s << 2) + khalf;
            v2f b;
            b.x = sW[k * HDIM + nt * 16 + n0];
            b.y = sW[(k + 1) * HDIM + nt * 16 + n0];
            c = __builtin_amdgcn_wmma_f32_16x16x4_f32(false, a[s], false, b,
                                                      (short)0, c, false, false);
        }
        const int colo = nt * 16 + n0;
#pragma unroll
        for (int r = 0; r < 8; ++r)
            out[(mb + r) * HDIM + colo] = c[r] + bf[colo];
    }
}

// ---------------------------------------------------------------- host
static inline long cdivl(long a, long b) { return (a + b - 1) / b; }

extern "C" void kernel_launch(void* const* d_in, const int* in_sizes, int n_in,
                              void* d_out, int out_size, void* d_ws, size_t ws_size,
                              hipStream_t stream) {
    const float* x     = (const float*)d_in[0];
    const int*   ei    = (const int*)d_in[1];
    const int*   batch = (const int*)d_in[2];
    const float* W[3]  = {(const float*)d_in[3], (const float*)d_in[7], (const float*)d_in[11]};
    const float* g[3]  = {(const float*)d_in[5], (const float*)d_in[9], (const float*)d_in[13]};
    const float* be[3] = {(const float*)d_in[6], (const float*)d_in[10], (const float*)d_in[14]};
    const float* fcw   = (const float*)d_in[15];
    const float* fcb   = (const float*)d_in[16];

    const int N = in_sizes[2];
    const int E = in_sizes[1] / 2;
    const int* rowI = ei;          // edge_index[0] = source
    const int* colI = ei + E;      // edge_index[1] = target

    float* ws    = (float*)d_ws;
    float* bufA  = ws;                                   // N*128 (hw)
    float* bufB  = bufA + (size_t)N * HDIM;              // N*128 (h)
    float* dinv  = bufB + (size_t)N * HDIM;              // N (deg -> rsqrt)
    float* sums  = dinv + N;                             // 128
    float* sumsq = sums + HDIM;                          // 128
    float* scale = sumsq + HDIM;                         // 128
    float* shift = scale + HDIM;                         // 128
    float* pool  = shift + HDIM;                         // 64*128
    float* cnt   = pool + GRAPHS * HDIM;                 // 64

    // -- GCN normalization prep (deg with self-loops -> dinv)
    k_fill<<<(unsigned)cdivl(N, 256), 256, 0, stream>>>(dinv, 1.0f, N);
    k_deg_accum<<<(unsigned)cdivl(E, 256), 256, 0, stream>>>(colI, dinv, E);
    k_rsqrt_inplace<<<(unsigned)cdivl(N, 256), 256, 0, stream>>>(dinv, N);

    const float* hin = x;
    const float invN = 1.0f / (float)N;
    for (int l = 0; l < 3; ++l) {
        k_gemm128<<<(unsigned)cdivl(N, 64), 128, 0, stream>>>(hin, W[l], bufA, N);
        // initialize scatter target with self-loop contribution (coalesced stores, no zero-fill)
        k_selfinit<<<(unsigned)cdivl((long)N * 32, 256), 256, 0, stream>>>(bufA, dinv, bufB, (long)N * 32);
        long sth = (long)E * 32;
        k_scatter<<<(unsigned)cdivl(sth, 256), 256, 0, stream>>>(bufA, rowI, colI, dinv, bufB, E);
        k_fill<<<1, 256, 0, stream>>>(sums, 0.f, 2 * HDIM);       // sums + sumsq contiguous
        k_stats<<<512, 256, 0, stream>>>(bufB, sums, sumsq, N);
        k_bn_finalize<<<1, 128, 0, stream>>>(sums, sumsq, g[l], be[l], scale, shift, invN);
        k_bnrelu<<<(unsigned)cdivl((long)N * 32, 256), 256, 0, stream>>>(bufB, scale, shift, (long)N * 32);
        hin = bufB;
    }

    k_fill<<<(unsigned)cdivl(GRAPHS * HDIM + GRAPHS, 256), 256, 0, stream>>>(pool, 0.f, GRAPHS * HDIM + GRAPHS);
    k_pool<<<256, 256, 0, stream>>>(bufB, batch, pool, cnt, N);
    k_fc<<<1, 128, 0, stream>>>(pool, cnt, fcw, fcb, (float*)d_out);
}